// PatchSampler_44092134261573
// MI455X (gfx1250) — compile-verified
//
#include <hip/hip_runtime.h>
#include <math.h>

// ---------------- problem constants ----------------
#define B    32
#define C    3
#define H    512
#define W    512
#define PS   64
#define K    32
#define STRIDE 16
#define GRID 29                 // (512-64)/16 + 1
#define N    (GRID*GRID)        // 841
#define TOTAL_U (B*N)           // 26912
#define HALF_U  (TOTAL_U/2)     // 13456

typedef float v2f __attribute__((ext_vector_type(2)));
typedef float v8f __attribute__((ext_vector_type(8)));

// ---------------- Kernel 1: avg-pool scores via WMMA f32 16x16x4 ----------------
// grid: (29, 32) = (y, batch). block: 256 threads (8 waves).
// Each wave task (y, cg): 16 chained v_wmma_f32_16x16x4_f32 with A = 16 cols x 4 rows
// of weights, B = ones -> D[m][*] = sum of 64 rows for column cg*16+m.
__global__ void ps_score_kernel(const float* __restrict__ weights,
                                float* __restrict__ scores) {
    __shared__ float s64[W];        // per-column 64-row sums for this output row y
    const int y   = blockIdx.x;     // 0..28
    const int b   = blockIdx.y;     // 0..31
    const int tid = threadIdx.x;
    const int wave = tid >> 5;
    const int lane = tid & 31;

    const float* Wb = weights + (size_t)b * H * W;
    const int m    = lane & 15;          // column within 16-group (M dim of A)
    const int r2   = (lane >> 4) * 2;    // lanes 16-31 hold K=2,3

    const v2f bones = {1.0f, 1.0f};

    for (int cg = wave; cg < 32; cg += 8) {
        const int cb = cg * 16;
        const float* colp = Wb + (size_t)(y * STRIDE + r2) * W + cb + m;
        v8f acc = {};
        #pragma unroll
        for (int j = 0; j < 16; ++j) {           // 16 * K=4 = 64 rows
            const float* p = colp + (size_t)j * 4 * W;
            v2f a;
            a.x = p[0];
            a.y = p[W];
            acc = __builtin_amdgcn_wmma_f32_16x16x4_f32(
                false, a, false, bones, (short)0, acc, false, false);
        }
        // D[m][n]: lane L<16 holds D[r][L] in acc[r]; lane L>=16 holds D[8+r][L-16].
        // All n identical -> lane 0 / lane 16 extract the 16 column sums.
        if (lane == 0) {
            #pragma unroll
            for (int r = 0; r < 8; ++r) s64[cb + r] = acc[r];
        } else if (lane == 16) {
            #pragma unroll
            for (int r = 0; r < 8; ++r) s64[cb + 8 + r] = acc[r];
        }
    }
    __syncthreads();

    // x-direction 64-tap box reduction, stride 16 -> 29 scores for this row
    if (tid < GRID) {
        const int xb = tid * STRIDE;
        float s = 0.0f;
        #pragma unroll 16
        for (int dx = 0; dx < PS; ++dx) s += s64[xb + dx];
        scores[(size_t)b * N + y * GRID + tid] = s * (1.0f / (PS * PS));
    }
}

// ---------------- threefry2x32-20 (exact JAX PRNG) ----------------
__device__ __forceinline__ unsigned rotl32(unsigned x, int r) {
    return (x << r) | (x >> (32 - r));
}
__device__ __forceinline__ void threefry2x32(unsigned k0, unsigned k1,
                                             unsigned x0, unsigned x1,
                                             unsigned& o0, unsigned& o1) {
    const unsigned ks[3] = {k0, k1, k0 ^ k1 ^ 0x1BD11BDAu};
    x0 += ks[0]; x1 += ks[1];
    const int rot[2][4] = {{13, 15, 26, 6}, {17, 29, 16, 24}};
    #pragma unroll
    for (int i = 0; i < 5; ++i) {
        const int* r = rot[i & 1];
        #pragma unroll
        for (int j = 0; j < 4; ++j) {
            x0 += x1; x1 = rotl32(x1, r[j]); x1 ^= x0;
        }
        x0 += ks[(i + 1) % 3];
        x1 += ks[(i + 2) % 3] + (unsigned)(i + 1);
    }
    o0 = x0; o1 = x1;
}

// ---------------- Kernel 2: normalize + Gumbel + top-K + coords ----------------
// grid: 32 (batch). block: 256.
__global__ void ps_topk_kernel(const float* __restrict__ scores,
                               int* __restrict__ ws_coords,
                               float* __restrict__ coords_out) {
    __shared__ float val[N];
    __shared__ float red[256];
    __shared__ int   redi[256];
    __shared__ float s_min, s_max;

    const int b   = blockIdx.x;
    const int tid = threadIdx.x;

    float lmin = INFINITY, lmax = -INFINITY;
    for (int n = tid; n < N; n += 256) {
        float s = scores[(size_t)b * N + n];
        val[n] = s;
        lmin = fminf(lmin, s);
        lmax = fmaxf(lmax, s);
    }
    red[tid] = lmin; __syncthreads();
    for (int off = 128; off > 0; off >>= 1) {
        if (tid < off) red[tid] = fminf(red[tid], red[tid + off]);
        __syncthreads();
    }
    if (tid == 0) s_min = red[0];
    __syncthreads();
    red[tid] = lmax; __syncthreads();
    for (int off = 128; off > 0; off >>= 1) {
        if (tid < off) red[tid] = fmaxf(red[tid], red[tid + off]);
        __syncthreads();
    }
    if (tid == 0) s_max = red[0];
    __syncthreads();

    const float inv_range = 1.0f / fmaxf(s_max - s_min, 1e-6f);
    for (int n = tid; n < N; n += 256) {
        float scaled = (val[n] - s_min) * inv_range * (1.0f / 0.3f);
        // exact JAX uniform(key(42), (32,841)): threefry over split counters
        unsigned i = (unsigned)(b * N + n);
        unsigned j = i % HALF_U;
        unsigned o0, o1;
        threefry2x32(0u, 42u, j, j + HALF_U, o0, o1);
        unsigned bits = (i < HALF_U) ? o0 : o1;
        float u = __uint_as_float((bits >> 9) | 0x3f800000u) - 1.0f;
        float g = -logf(-logf(u + 1e-20f) + 1e-20f);
        val[n] = scaled + g;
    }
    __syncthreads();

    // 32 rounds of argmax (tie-break: lowest index), matching lax.top_k order
    for (int k = 0; k < K; ++k) {
        float bv = -INFINITY; int bi = 0x7fffffff;
        for (int n = tid; n < N; n += 256) {
            float v = val[n];
            if (v > bv || (v == bv && n < bi)) { bv = v; bi = n; }
        }
        red[tid] = bv; redi[tid] = bi;
        __syncthreads();
        for (int off = 128; off > 0; off >>= 1) {
            if (tid < off) {
                float v2 = red[tid + off]; int i2 = redi[tid + off];
                if (v2 > red[tid] || (v2 == red[tid] && i2 < redi[tid])) {
                    red[tid] = v2; redi[tid] = i2;
                }
            }
            __syncthreads();
        }
        if (tid == 0) {
            int idx = redi[0];
            int hh = (idx / GRID) * STRIDE; if (hh > H - PS) hh = H - PS;
            int ww = (idx % GRID) * STRIDE; if (ww > W - PS) ww = W - PS;
            int o = (b * K + k) * 2;
            ws_coords[o + 0] = hh;
            ws_coords[o + 1] = ww;
            coords_out[o + 0] = (float)hh;
            coords_out[o + 1] = (float)ww;
            val[idx] = -INFINITY;
        }
        __syncthreads();
    }
}

// ---------------- Kernel 3: patch gather (128-bit loads/stores) ----------------
// grid: B*K = 1024. block: 256. Copies 3 image channels + 1 label channel.
__global__ void ps_gather_kernel(const float* __restrict__ image,
                                 const float* __restrict__ labels,
                                 const int* __restrict__ ws_coords,
                                 float* __restrict__ patches,
                                 float* __restrict__ patch_labels) {
    const int bk  = blockIdx.x;          // b*K + k
    const int b   = bk >> 5;
    const int tid = threadIdx.x;
    const int hh  = ws_coords[bk * 2 + 0];
    const int ww  = ws_coords[bk * 2 + 1];

    // 3 image channels
    #pragma unroll
    for (int c = 0; c < C; ++c) {
        const float* src = image + ((size_t)(b * C + c) * H + hh) * W + ww;
        float4* dst4 = (float4*)(patches + (size_t)(bk * C + c) * (PS * PS));
        for (int q = tid; q < (PS * PS) / 4; q += 256) {   // 1024 float4
            const int row = q >> 4;        // 16 float4 per 64-float row
            const int c4  = q & 15;
            const float* sp = src + (size_t)row * W + c4 * 4;
            __builtin_prefetch(sp + 4 * W, 0, 1);          // global_prefetch_b8
            dst4[q] = *(const float4*)sp;
        }
    }
    // 1 label channel
    {
        const float* src = labels + ((size_t)b * H + hh) * W + ww;
        float4* dst4 = (float4*)(patch_labels + (size_t)bk * (PS * PS));
        for (int q = tid; q < (PS * PS) / 4; q += 256) {
            const int row = q >> 4;
            const int c4  = q & 15;
            dst4[q] = *(const float4*)(src + (size_t)row * W + c4 * 4);
        }
    }
}

// ---------------- launch ----------------
extern "C" void kernel_launch(void* const* d_in, const int* in_sizes, int n_in,
                              void* d_out, int out_size, void* d_ws, size_t ws_size,
                              hipStream_t stream) {
    const float* image   = (const float*)d_in[0];
    const float* labels  = (const float*)d_in[1];
    const float* weights = (const float*)d_in[2];
    float* out = (float*)d_out;

    // output layout: patches | patch_labels | coords
    const size_t patchesN = (size_t)B * K * C * PS * PS;   // 12,582,912
    const size_t labelsN  = (size_t)B * K * PS * PS;       //  4,194,304
    float* out_patches = out;
    float* out_labels  = out + patchesN;
    float* out_coords  = out + patchesN + labelsN;

    // workspace: scores (26912 f32) then int coords (2048 i32)
    float* ws_scores = (float*)d_ws;
    int*   ws_coords = (int*)((char*)d_ws + 131072);

    dim3 g1(GRID, B);
    ps_score_kernel<<<g1, 256, 0, stream>>>(weights, ws_scores);
    ps_topk_kernel<<<B, 256, 0, stream>>>(ws_scores, ws_coords, out_coords);
    ps_gather_kernel<<<B * K, 256, 0, stream>>>(image, labels, ws_coords,
                                                out_patches, out_labels);
}